// Vectorizer_65025804861835
// MI455X (gfx1250) — compile-verified
//
#include <hip/hip_runtime.h>

// Soft-argmax over K=7 channel groups, NCHW layout.
// B=4, C=14 (=OC 2 * K 7), H=W=1024.  Pure HBM-streaming kernel:
//   235 MB in + 33.5 MB out @ 23.3 TB/s  ->  ~11.5 us floor.  AI ~1 FLOP/byte
// -> matrix pipes irrelevant; the MI455X-specific wins are on the memory path:
//   * global_load_b128 / global_store_b128, fully coalesced per plane
//   * CDNA5 TH=NT cache hints (input > 192 MB L2, every byte touched once)
//   * scalar (SADDR) base addressing: batch index lives in blockIdx.y so the
//     per-batch base pointer is wave-uniform -> SGPR base + 32-bit voffset,
//     no 64-bit VGPR carry chains per plane.

typedef float f32x4 __attribute__((ext_vector_type(4)));

#define HW_   (1024 * 1024)
#define QPI_  (HW_ / 4)        // float4 "quads" per image plane = 2^18
#define NCH_  14               // OUT_CHANNELS * COMPONENT_CHANNELS
#define KCH_  7

static __device__ __forceinline__ f32x4 exp4(f32x4 v) {
    f32x4 r;
    r.x = __expf(v.x);
    r.y = __expf(v.y);
    r.z = __expf(v.z);
    r.w = __expf(v.w);
    return r;
}

static __device__ __forceinline__ f32x4 rcp4(f32x4 v) {
    f32x4 r;
    r.x = __builtin_amdgcn_rcpf(v.x);
    r.y = __builtin_amdgcn_rcpf(v.y);
    r.z = __builtin_amdgcn_rcpf(v.z);
    r.w = __builtin_amdgcn_rcpf(v.w);
    return r;
}

__global__ __launch_bounds__(256) void softargmax7_kernel(
    const float* __restrict__ x, float* __restrict__ out) {
    // blockIdx.y = batch (wave-uniform -> SGPR base), x-dim covers one plane.
    const int b = blockIdx.y;
    const int q = blockIdx.x * blockDim.x + threadIdx.x;   // float4 quad in plane

    const f32x4* __restrict__ xb =
        reinterpret_cast<const f32x4*>(x) + (size_t)b * NCH_ * QPI_;
    f32x4* __restrict__ ob =
        reinterpret_cast<f32x4*>(out) + (size_t)b * 2 * QPI_;

    // Issue all 14 coalesced b128 NT loads up front (VMEM burst, one wait).
    f32x4 v[NCH_];
#pragma unroll
    for (int c = 0; c < NCH_; ++c)
        v[c] = __builtin_nontemporal_load(xb + (size_t)c * QPI_ + q);

#pragma unroll
    for (int o = 0; o < 2; ++o) {
        const f32x4* g = &v[o * KCH_];

        // per-lane (4 pixels) max over K=7, matching jax.nn.softmax
        f32x4 m = g[0];
#pragma unroll
        for (int k = 1; k < KCH_; ++k)
            m = __builtin_elementwise_max(m, g[k]);

        f32x4 den = {0.f, 0.f, 0.f, 0.f};
        f32x4 num = {0.f, 0.f, 0.f, 0.f};
#pragma unroll
        for (int k = 0; k < KCH_; ++k) {
            f32x4 e = exp4(g[k] - m);
            den = den + e;
            const float w = (float)(k - KCH_ / 2);   // -3..3, w==0 at k=3
            if (k != KCH_ / 2)
                num = num + w * e;
        }

        f32x4 r = num * rcp4(den);   // v_rcp_f32: ~1 ulp, fine for softmax norm
        __builtin_nontemporal_store(r, ob + (size_t)o * QPI_ + q);
    }
}

extern "C" void kernel_launch(void* const* d_in, const int* in_sizes, int n_in,
                              void* d_out, int out_size, void* d_ws, size_t ws_size,
                              hipStream_t stream) {
    (void)n_in; (void)out_size; (void)d_ws; (void)ws_size;
    const float* x = (const float*)d_in[0];
    float*     out = (float*)d_out;

    const int B = in_sizes[0] / (NCH_ * HW_);    // 4
    dim3 block(256, 1, 1);
    dim3 grid(QPI_ / 256, B, 1);                 // 1024 x 4 blocks

    softargmax7_kernel<<<grid, block, 0, stream>>>(x, out);
}